// EthereumGraphSAGE_317827579953
// MI455X (gfx1250) — compile-verified
//
#include <hip/hip_runtime.h>

typedef __attribute__((ext_vector_type(2))) float v2f;
typedef __attribute__((ext_vector_type(8))) float v8f;

#define WMMA_F32(a, b, c)                                                     \
  __builtin_amdgcn_wmma_f32_16x16x4_f32(false, (a), false, (b), (short)0,     \
                                        (c), false, false)

static inline int imin_host(int a, int b) { return a < b ? a : b; }

// ---------------------------------------------------------------------------
// h_G = graph_attr @ W_graph + b_graph   (tiny: [1,16] x [16,16])
// ---------------------------------------------------------------------------
__global__ void graph_lin_kernel(const float* __restrict__ ga,
                                 const float* __restrict__ Wg,
                                 const float* __restrict__ bg,
                                 float* __restrict__ hG) {
  int j = threadIdx.x;
  if (j < 16) {
    float s = bg[j];
#pragma unroll
    for (int k = 0; k < 16; ++k) s += ga[k] * Wg[k * 16 + j];
    hG[j] = s;
  }
}

// ---------------------------------------------------------------------------
// h_i = x @ W_node + b_node   ([N,32] x [32,32]) via V_WMMA_F32_16X16X4_F32
// One wave per 16-node tile (grid-stride). 8 K-steps, 2 N-tiles.
// ---------------------------------------------------------------------------
__global__ void node_lin_kernel(const float* __restrict__ x,
                                const float* __restrict__ W,   // [32,32]
                                const float* __restrict__ b,   // [32]
                                float* __restrict__ h_i, int N) {
  const int lane = threadIdx.x & 31;
  const int wid = blockIdx.x * (blockDim.x >> 5) + (threadIdx.x >> 5);
  const int nwaves = gridDim.x * (blockDim.x >> 5);
  const int n = lane & 15;
  const int khalf = (lane >> 4) * 2;
  const int moff = (lane >> 4) * 8;

  v2f bf0[8], bf1[8];
#pragma unroll
  for (int ks = 0; ks < 8; ++ks) {
    int k = ks * 4 + khalf;
    bf0[ks].x = W[(k + 0) * 32 + n];
    bf0[ks].y = W[(k + 1) * 32 + n];
    bf1[ks].x = W[(k + 0) * 32 + 16 + n];
    bf1[ks].y = W[(k + 1) * 32 + 16 + n];
  }
  const float bias0 = b[n];
  const float bias1 = b[16 + n];

  const int ntiles = (N + 15) >> 4;
  for (int tile = wid; tile < ntiles; tile += nwaves) {
    const int base = tile << 4;
    int m = base + n;
    if (m >= N) m = N - 1;
    v8f acc0 = {}, acc1 = {};
#pragma unroll
    for (int ks = 0; ks < 8; ++ks) {
      int k = ks * 4 + khalf;
      v2f a = *reinterpret_cast<const v2f*>(x + (size_t)m * 32 + k);
      acc0 = WMMA_F32(a, bf0[ks], acc0);
      acc1 = WMMA_F32(a, bf1[ks], acc1);
    }
    if (base + 16 <= N) {  // uniform full-tile fast path (always taken here)
#pragma unroll
      for (int r = 0; r < 8; ++r) {
        const int node = base + r + moff;
        h_i[(size_t)node * 32 + n] = acc0[r] + bias0;
        h_i[(size_t)node * 32 + 16 + n] = acc1[r] + bias1;
      }
    } else {
#pragma unroll
      for (int r = 0; r < 8; ++r) {
        const int node = base + r + moff;
        if (node < N) {
          h_i[(size_t)node * 32 + n] = acc0[r] + bias0;
          h_i[(size_t)node * 32 + 16 + n] = acc1[r] + bias1;
        }
      }
    }
  }
}

// ---------------------------------------------------------------------------
// Edge aggregator: h_e = relu(concat(h_i[row], h_i[col], edge_attr, h_G) @ W_e + b_e)
// scatter-add into m_sum[row] + degree counts.  [E,96] x [96,16] via WMMA f32.
// Two independent accumulator chains (even/odd K-steps) for matrix-pipe ILP.
// ---------------------------------------------------------------------------
__global__ void edge_agg_kernel(const float* __restrict__ h_i,
                                const float* __restrict__ edge_attr,
                                const float* __restrict__ hG,
                                const int* __restrict__ row,
                                const int* __restrict__ col,
                                const float* __restrict__ We,  // [96,16]
                                const float* __restrict__ be,  // [16]
                                float* __restrict__ m_sum,     // [N,16]
                                float* __restrict__ deg_row,   // [N]
                                int E) {
  const int lane = threadIdx.x & 31;
  const int wid = blockIdx.x * (blockDim.x >> 5) + (threadIdx.x >> 5);
  const int nwaves = gridDim.x * (blockDim.x >> 5);
  const int n = lane & 15;
  const int khalf = (lane >> 4) * 2;
  const int moff = (lane >> 4) * 8;

  // Hoisted B fragments: 24 K-steps of W_edge_agg.
  v2f bf[24];
#pragma unroll
  for (int ks = 0; ks < 24; ++ks) {
    int k = ks * 4 + khalf;
    bf[ks].x = We[(k + 0) * 16 + n];
    bf[ks].y = We[(k + 1) * 16 + n];
  }
  const float bias = be[n];
  // h_G A-fragments (K 80..95) are identical for every row M.
  v2f ahg[4];
#pragma unroll
  for (int ks = 0; ks < 4; ++ks) {
    int k = ks * 4 + khalf;
    ahg[ks].x = hG[k];
    ahg[ks].y = hG[k + 1];
  }

  const int ntiles = (E + 15) >> 4;
  for (int tile = wid; tile < ntiles; tile += nwaves) {
    const int base = tile << 4;
    int e = base + n;
    const bool live = (e < E);
    if (!live) e = E - 1;
    const int r0 = row[e];
    const int c0 = col[e];

    v8f accA = {}, accB = {};
#pragma unroll
    for (int ks = 0; ks < 8; ++ks) {  // K 0..31 : h_i[row]
      int k = ks * 4 + khalf;
      v2f a = *reinterpret_cast<const v2f*>(h_i + (size_t)r0 * 32 + k);
      if (ks & 1) accB = WMMA_F32(a, bf[ks], accB);
      else        accA = WMMA_F32(a, bf[ks], accA);
    }
#pragma unroll
    for (int ks = 0; ks < 8; ++ks) {  // K 32..63 : h_i[col]
      int k = ks * 4 + khalf;
      v2f a = *reinterpret_cast<const v2f*>(h_i + (size_t)c0 * 32 + k);
      if (ks & 1) accB = WMMA_F32(a, bf[8 + ks], accB);
      else        accA = WMMA_F32(a, bf[8 + ks], accA);
    }
#pragma unroll
    for (int ks = 0; ks < 4; ++ks) {  // K 64..79 : edge_attr
      int k = ks * 4 + khalf;
      v2f a = *reinterpret_cast<const v2f*>(edge_attr + (size_t)e * 16 + k);
      if (ks & 1) accB = WMMA_F32(a, bf[16 + ks], accB);
      else        accA = WMMA_F32(a, bf[16 + ks], accA);
    }
#pragma unroll
    for (int ks = 0; ks < 4; ++ks) {  // K 80..95 : h_G (constant fragments)
      if (ks & 1) accB = WMMA_F32(ahg[ks], bf[20 + ks], accB);
      else        accA = WMMA_F32(ahg[ks], bf[20 + ks], accA);
    }

    // Scatter: lane L, vgpr r -> edge (base + r + 8*(L/16)), channel n.
    if (base + 16 <= E) {  // uniform full-tile fast path (always taken here)
#pragma unroll
      for (int r = 0; r < 8; ++r) {
        const int eM = base + r + moff;
        float v = accA[r] + accB[r] + bias;
        v = v > 0.0f ? v : 0.0f;
        const int dst = row[eM];
        atomicAdd(&m_sum[(size_t)dst * 16 + n], v);
      }
      if (lane < 16) atomicAdd(&deg_row[r0], 1.0f);
    } else {
#pragma unroll
      for (int r = 0; r < 8; ++r) {
        const int eM = base + r + moff;
        if (eM < E) {
          float v = accA[r] + accB[r] + bias;
          v = v > 0.0f ? v : 0.0f;
          const int dst = row[eM];
          atomicAdd(&m_sum[(size_t)dst * 16 + n], v);
        }
      }
      if (lane < 16 && live) atomicAdd(&deg_row[r0], 1.0f);
    }
  }
}

// ---------------------------------------------------------------------------
// h_i2 = relu(concat(h_i, m_sum/deg, h_G) @ W_node_agg + b)  ([N,64]x[64,32])
// ---------------------------------------------------------------------------
__global__ void node_agg_kernel(const float* __restrict__ h_i,
                                const float* __restrict__ m_sum,
                                const float* __restrict__ deg_row,
                                const float* __restrict__ hG,
                                const float* __restrict__ Wn,  // [64,32]
                                const float* __restrict__ bn,  // [32]
                                float* __restrict__ h_i2, int N) {
  const int lane = threadIdx.x & 31;
  const int wid = blockIdx.x * (blockDim.x >> 5) + (threadIdx.x >> 5);
  const int nwaves = gridDim.x * (blockDim.x >> 5);
  const int n = lane & 15;
  const int khalf = (lane >> 4) * 2;
  const int moff = (lane >> 4) * 8;

  v2f bf0[16], bf1[16];
#pragma unroll
  for (int ks = 0; ks < 16; ++ks) {
    int k = ks * 4 + khalf;
    bf0[ks].x = Wn[(k + 0) * 32 + n];
    bf0[ks].y = Wn[(k + 1) * 32 + n];
    bf1[ks].x = Wn[(k + 0) * 32 + 16 + n];
    bf1[ks].y = Wn[(k + 1) * 32 + 16 + n];
  }
  const float bias0 = bn[n];
  const float bias1 = bn[16 + n];
  v2f ahg[4];
#pragma unroll
  for (int ks = 0; ks < 4; ++ks) {
    int k = ks * 4 + khalf;
    ahg[ks].x = hG[k];
    ahg[ks].y = hG[k + 1];
  }

  const int ntiles = (N + 15) >> 4;
  for (int tile = wid; tile < ntiles; tile += nwaves) {
    const int base = tile << 4;
    int m = base + n;
    if (m >= N) m = N - 1;
    const float invd = 1.0f / fmaxf(deg_row[m], 1.0f);

    v8f acc0 = {}, acc1 = {};
#pragma unroll
    for (int ks = 0; ks < 8; ++ks) {  // K 0..31 : h_i
      int k = ks * 4 + khalf;
      v2f a = *reinterpret_cast<const v2f*>(h_i + (size_t)m * 32 + k);
      acc0 = WMMA_F32(a, bf0[ks], acc0);
      acc1 = WMMA_F32(a, bf1[ks], acc1);
    }
#pragma unroll
    for (int ks = 0; ks < 4; ++ks) {  // K 32..47 : mean of edge messages
      int k = ks * 4 + khalf;
      v2f a = *reinterpret_cast<const v2f*>(m_sum + (size_t)m * 16 + k);
      a.x *= invd;
      a.y *= invd;
      acc0 = WMMA_F32(a, bf0[8 + ks], acc0);
      acc1 = WMMA_F32(a, bf1[8 + ks], acc1);
    }
#pragma unroll
    for (int ks = 0; ks < 4; ++ks) {  // K 48..63 : h_G
      acc0 = WMMA_F32(ahg[ks], bf0[12 + ks], acc0);
      acc1 = WMMA_F32(ahg[ks], bf1[12 + ks], acc1);
    }
    if (base + 16 <= N) {  // uniform full-tile fast path (always taken here)
#pragma unroll
      for (int r = 0; r < 8; ++r) {
        const int node = base + r + moff;
        float v0 = acc0[r] + bias0;
        float v1 = acc1[r] + bias1;
        h_i2[(size_t)node * 32 + n] = v0 > 0.0f ? v0 : 0.0f;
        h_i2[(size_t)node * 32 + 16 + n] = v1 > 0.0f ? v1 : 0.0f;
      }
    } else {
#pragma unroll
      for (int r = 0; r < 8; ++r) {
        const int node = base + r + moff;
        if (node < N) {
          float v0 = acc0[r] + bias0;
          float v1 = acc1[r] + bias1;
          h_i2[(size_t)node * 32 + n] = v0 > 0.0f ? v0 : 0.0f;
          h_i2[(size_t)node * 32 + 16 + n] = v1 > 0.0f ? v1 : 0.0f;
        }
      }
    }
  }
}

// ---------------------------------------------------------------------------
// SAGE mean-aggregation scatter: agg_sum[col] += h_i2[row]; deg_col[col] += 1
// One thread per (edge, 4-channel chunk): float4 load + 4 atomics.
// ---------------------------------------------------------------------------
__global__ void sage_scatter_kernel(const float* __restrict__ h_i2,
                                    const int* __restrict__ row,
                                    const int* __restrict__ col,
                                    float* __restrict__ agg_sum,
                                    float* __restrict__ deg_col, int E) {
  const long total = (long)E * 8;
  const long stride = (long)gridDim.x * blockDim.x;
  for (long t = (long)blockIdx.x * blockDim.x + threadIdx.x; t < total;
       t += stride) {
    const int e = (int)(t >> 3);
    const int c = (int)(t & 7);
    const int r = row[e];
    const int cl = col[e];
    const float4 v =
        *reinterpret_cast<const float4*>(h_i2 + (size_t)r * 32 + c * 4);
    float* dst = agg_sum + (size_t)cl * 32 + c * 4;
    atomicAdd(dst + 0, v.x);
    atomicAdd(dst + 1, v.y);
    atomicAdd(dst + 2, v.z);
    atomicAdd(dst + 3, v.w);
    if (c == 0) atomicAdd(&deg_col[cl], 1.0f);
  }
}

// ---------------------------------------------------------------------------
// h_nodes = (agg_sum/deg) @ W_sage_l + b + h_i2 @ W_sage_r, then mean-pool.
// Block-local LDS reduction, then 8 global atomics per block.
// ---------------------------------------------------------------------------
__global__ void final_pool_kernel(const float* __restrict__ h_i2,
                                  const float* __restrict__ agg_sum,
                                  const float* __restrict__ deg_col,
                                  const float* __restrict__ Wl,  // [32,8]
                                  const float* __restrict__ bl,  // [8]
                                  const float* __restrict__ Wr,  // [32,8]
                                  float* __restrict__ pool, int N) {
  __shared__ float sWl[256], sWr[256], sbl[8], sacc[8];
  const int tid = threadIdx.x;
  for (int i = tid; i < 256; i += blockDim.x) {
    sWl[i] = Wl[i];
    sWr[i] = Wr[i];
  }
  if (tid < 8) {
    sbl[tid] = bl[tid];
    sacc[tid] = 0.0f;
  }
  __syncthreads();

  float lacc[8];
#pragma unroll
  for (int j = 0; j < 8; ++j) lacc[j] = 0.0f;

  const int stride = gridDim.x * blockDim.x;
  for (int i = blockIdx.x * blockDim.x + tid; i < N; i += stride) {
    const float invd = 1.0f / fmaxf(deg_col[i], 1.0f);
    float o[8];
#pragma unroll
    for (int j = 0; j < 8; ++j) o[j] = sbl[j];
#pragma unroll 4
    for (int k = 0; k < 32; ++k) {
      const float a = agg_sum[(size_t)i * 32 + k] * invd;
      const float h = h_i2[(size_t)i * 32 + k];
#pragma unroll
      for (int j = 0; j < 8; ++j) o[j] += a * sWl[k * 8 + j] + h * sWr[k * 8 + j];
    }
#pragma unroll
    for (int j = 0; j < 8; ++j) lacc[j] += o[j];
  }
#pragma unroll
  for (int j = 0; j < 8; ++j) atomicAdd(&sacc[j], lacc[j]);
  __syncthreads();
  if (tid < 8) atomicAdd(&pool[tid], sacc[tid]);
}

// ---------------------------------------------------------------------------
// log_softmax over the pooled [1,8] vector.
// ---------------------------------------------------------------------------
__global__ void softmax_kernel(const float* __restrict__ pool,
                               float* __restrict__ out, int N) {
  if (threadIdx.x == 0 && blockIdx.x == 0) {
    float p[8];
    float mx = -3.4e38f;
#pragma unroll
    for (int j = 0; j < 8; ++j) {
      p[j] = pool[j] / (float)N;
      mx = fmaxf(mx, p[j]);
    }
    float s = 0.0f;
#pragma unroll
    for (int j = 0; j < 8; ++j) s += expf(p[j] - mx);
    const float ls = logf(s);
#pragma unroll
    for (int j = 0; j < 8; ++j) out[j] = p[j] - mx - ls;
  }
}

// ---------------------------------------------------------------------------
extern "C" void kernel_launch(void* const* d_in, const int* in_sizes, int n_in,
                              void* d_out, int out_size, void* d_ws,
                              size_t ws_size, hipStream_t stream) {
  const float* x = (const float*)d_in[0];
  const float* edge_attr = (const float*)d_in[1];
  const float* graph_attr = (const float*)d_in[2];
  const int* edge_index = (const int*)d_in[3];
  // d_in[4] = batch (all zeros; single-graph pooling handled explicitly)
  const float* W_node = (const float*)d_in[5];
  const float* b_node = (const float*)d_in[6];
  const float* W_graph = (const float*)d_in[7];
  const float* b_graph = (const float*)d_in[8];
  const float* W_edge = (const float*)d_in[9];
  const float* b_edge = (const float*)d_in[10];
  const float* W_nagg = (const float*)d_in[11];
  const float* b_nagg = (const float*)d_in[12];
  const float* W_sl = (const float*)d_in[13];
  const float* b_sl = (const float*)d_in[14];
  const float* W_sr = (const float*)d_in[15];

  const int N = in_sizes[0] / 32;
  const int E = in_sizes[1] / 16;
  const int* row = edge_index;
  const int* col = edge_index + E;

  float* ws = (float*)d_ws;
  size_t off = 0;
  // -------- region that must be zeroed each launch --------
  float* pool = ws + off;     off += 32;
  float* m_sum = ws + off;    off += (size_t)N * 16;
  float* deg_row = ws + off;  off += (size_t)N;
  float* agg_sum = ws + off;  off += (size_t)N * 32;
  float* deg_col = ws + off;  off += (size_t)N;
  const size_t zero_floats = off;
  // -------- write-before-read region --------
  float* hG = ws + off;       off += 16;
  float* h_i = ws + off;      off += (size_t)N * 32;
  float* h_i2 = ws + off;     off += (size_t)N * 32;

  hipMemsetAsync(d_ws, 0, zero_floats * sizeof(float), stream);

  graph_lin_kernel<<<1, 32, 0, stream>>>(graph_attr, W_graph, b_graph, hG);

  const int ntilesN = (N + 15) >> 4;
  const int blocks_node = imin_host((ntilesN + 7) / 8, 1024);
  node_lin_kernel<<<blocks_node, 256, 0, stream>>>(x, W_node, b_node, h_i, N);

  const int ntilesE = (E + 15) >> 4;
  const int blocks_edge = imin_host((ntilesE + 7) / 8, 2048);
  edge_agg_kernel<<<blocks_edge, 256, 0, stream>>>(
      h_i, edge_attr, hG, row, col, W_edge, b_edge, m_sum, deg_row, E);

  node_agg_kernel<<<blocks_node, 256, 0, stream>>>(h_i, m_sum, deg_row, hG,
                                                   W_nagg, b_nagg, h_i2, N);

  sage_scatter_kernel<<<2048, 256, 0, stream>>>(h_i2, row, col, agg_sum,
                                                deg_col, E);

  final_pool_kernel<<<1024, 256, 0, stream>>>(h_i2, agg_sum, deg_col, W_sl,
                                              b_sl, W_sr, pool, N);

  softmax_kernel<<<1, 32, 0, stream>>>(pool, (float*)d_out, N);
}